// MultiHeadAttention_61546881352486
// MI455X (gfx1250) — compile-verified
//
#include <hip/hip_runtime.h>
#include <math.h>

// ---------------------------------------------------------------------------
// MI455X / gfx1250 wave32 WMMA implementation of causal MHA + RoPE.
// - V_WMMA_F32_16X16X4_F32 (full fp32 matrix pipe) for all matmuls
// - GLOBAL_LOAD_ASYNC_TO_LDS_B128 (ASYNCcnt) for GEMM B-tile staging
// - Non-temporal hints on the 512MB attn_probs stream (protect 192MB L2)
// Workspace requirement: 96 MB (qkv 48MB + q/k/v 3*16MB; attn_out aliases qkv).
// ---------------------------------------------------------------------------

typedef float v2f __attribute__((ext_vector_type(2)));
typedef float v4f __attribute__((ext_vector_type(4)));
typedef float v8f __attribute__((ext_vector_type(8)));

__device__ __forceinline__ v8f wmma_f32x4(v2f a, v2f b, v8f c) {
  // D = A(16x4 f32) * B(4x16 f32) + C(16x16 f32)
  return __builtin_amdgcn_wmma_f32_16x16x4_f32(false, a, false, b, (short)0, c,
                                               false, false);
}

// gfx1250 async DMA: global -> LDS, 16 bytes per lane, tracked by ASYNCcnt.
__device__ __forceinline__ void async_load_b128_to_lds(unsigned lds_off,
                                                       const float* gaddr) {
  asm volatile("global_load_async_to_lds_b128 %0, %1, off"
               :
               : "v"(lds_off), "v"((unsigned long long)(uintptr_t)gaddr)
               : "memory");
}
__device__ __forceinline__ void wait_async0() {
  asm volatile("s_wait_asynccnt 0x0" ::: "memory");
}

// ---------------------------------------------------------------------------
// Generic C(M,N) = A(M,K) @ B(N,K)^T   (row-major, M,N mult of 64, K mult 32).
// Block = 128 threads (4 waves); block tile 64x64; wave w owns rows
// [w*16, w*16+16) x all 64 cols (4 WMMA accumulators). B tile staged in LDS
// via async DMA, padded stride 36 dwords for conflict-free ds_load_b64 frags.
// ---------------------------------------------------------------------------
__global__ __launch_bounds__(128) void gemm_nt_kernel(
    const float* __restrict__ A, const float* __restrict__ B,
    float* __restrict__ C, int M, int N, int K) {
  __shared__ float bs[64 * 36];  // [ncol][k], stride 36 floats
  const int tid  = threadIdx.x;
  const int wave = tid >> 5;
  const int lane = tid & 31;
  const int l16  = lane & 15;
  const int hi   = lane >> 4;  // 0: lanes 0-15, 1: lanes 16-31
  const int n0   = blockIdx.x * 64;
  const int m0   = blockIdx.y * 64;
  const int mrow = m0 + wave * 16 + l16;  // A row this lane supplies

  v8f zero = {0.f, 0.f, 0.f, 0.f, 0.f, 0.f, 0.f, 0.f};
  v8f acc[4];
#pragma unroll
  for (int t = 0; t < 4; ++t) acc[t] = zero;

  for (int k0 = 0; k0 < K; k0 += 32) {
    // Async-stage B[n0..n0+63][k0..k0+31] into LDS (16B per lane per issue).
#pragma unroll
    for (int it = 0; it < 4; ++it) {
      int idx = tid + it * 128;   // 0..511 float4 slots
      int row = idx >> 3;         // 0..63
      int c4  = (idx & 7) * 4;    // 0,4,..,28
      async_load_b128_to_lds((unsigned)(uintptr_t)(bs + row * 36 + c4),
                             B + (size_t)(n0 + row) * K + k0 + c4);
    }
    wait_async0();
    __syncthreads();
#pragma unroll
    for (int ki = 0; ki < 32; ki += 4) {
      const int kk = ki + hi * 2;  // f32 A layout: v0=K(hi*2), v1=K(hi*2+1)
      v2f a = *(const v2f*)(A + (size_t)mrow * K + k0 + kk);
#pragma unroll
      for (int t = 0; t < 4; ++t) {
        v2f b = *(const v2f*)(bs + (t * 16 + l16) * 36 + kk);
        acc[t] = wmma_f32x4(a, b, acc[t]);
      }
    }
    __syncthreads();
  }

  // C/D layout: VGPR r -> row (r + hi*8), col = l16
  const int crow0 = m0 + wave * 16 + hi * 8;
#pragma unroll
  for (int t = 0; t < 4; ++t) {
    float* cp = C + (size_t)crow0 * N + n0 + t * 16 + l16;
#pragma unroll
    for (int r = 0; r < 8; ++r) cp[(size_t)r * N] = acc[t][r];
  }
}

// ---------------------------------------------------------------------------
// RoPE + head split:  qkv (B*T, 3D) -> q,k,v in (B,H,T,d_k), RoPE on q,k.
// One thread per (b,t,h,i<32) rotation pair. 2^21 threads total.
// ---------------------------------------------------------------------------
__global__ __launch_bounds__(256) void rope_split_kernel(
    const float* __restrict__ qkv, float* __restrict__ q,
    float* __restrict__ k, float* __restrict__ v) {
  const unsigned gid = blockIdx.x * 256u + threadIdx.x;  // < 2^21
  const int i = gid & 31;
  const int h = (gid >> 5) & 15;
  const int t = (gid >> 9) & 2047;
  const int b = gid >> 20;

  const size_t src = (size_t)(b * 2048 + t) * 3072 + h * 64 + i;
  const size_t dst = (size_t)((b * 16 + h) * 2048 + t) * 64 + i;

  // inv_freq = 10000^(-i/32) = 2^(-i*log2(10000)/32)
  const float inv = exp2f(-(float)i * 0.4152410118609203f);
  const float ang = (float)t * inv;
  float sv, cv;
  sincosf(ang, &sv, &cv);

  float q1 = qkv[src],        q2 = qkv[src + 32];
  float k1 = qkv[src + 1024], k2 = qkv[src + 1024 + 32];
  float v1 = qkv[src + 2048], v2 = qkv[src + 2048 + 32];

  q[dst]      = q1 * cv - q2 * sv;
  q[dst + 32] = q1 * sv + q2 * cv;
  k[dst]      = k1 * cv - k2 * sv;
  k[dst + 32] = k1 * sv + k2 * cv;
  v[dst]      = v1;
  v[dst + 32] = v2;
}

// ---------------------------------------------------------------------------
// Zero the attn_probs region (upper triangle must be exactly 0). NT stores.
// ---------------------------------------------------------------------------
__global__ __launch_bounds__(256) void zero_f4_kernel(v4f* __restrict__ p) {
  size_t idx = (size_t)blockIdx.x * 256 + threadIdx.x;
  v4f z = {0.f, 0.f, 0.f, 0.f};
  __builtin_nontemporal_store(z, &p[idx]);
}

// ---------------------------------------------------------------------------
// Causal attention for one (b,h): scores = q@k^T/8, softmax, probs out,
// attn_out = probs@v (written in (B,T,H*dk) layout for the output GEMM).
// Grid: (T/64, B*H), block 128 (4 waves); wave owns 16 query rows.
//   Pass 1: QK WMMAs -> row max (shfl_xor reduce within 16-lane groups)
//   Pass 2: QK again; e=exp(s-max); NT-stream e to probs; rowsum; e@V WMMAs
//           (P goes C-layout -> LDS -> A-layout, wave-private buffer)
//   Pass 3: scale attn_out by 1/rowsum; NT-rescale probs in place.
// ---------------------------------------------------------------------------
__global__ __launch_bounds__(128) void attn_kernel(
    const float* __restrict__ Q, const float* __restrict__ K,
    const float* __restrict__ V, float* __restrict__ probs,
    float* __restrict__ out, int T, int H) {
  __shared__ float pbuf[4][16 * 18];  // per-wave 16x16 P tile, stride 18
  const int tid  = threadIdx.x;
  const int wave = tid >> 5;
  const int lane = tid & 31;
  const int l16  = lane & 15;
  const int hi   = lane >> 4;
  const int bh = blockIdx.y;
  const int b  = bh / H;
  const int h  = bh % H;
  const int q0 = blockIdx.x * 64 + wave * 16;  // first query row of this wave

  const float* qp = Q + (size_t)bh * T * 64;
  const float* kp = K + (size_t)bh * T * 64;
  const float* vp = V + (size_t)bh * T * 64;
  float* pp = probs + (size_t)bh * T * T;

  // Q fragments: 16 x (16 queries x 4 d)  -> qa[f], f covers d = 4f..4f+3
  v2f qa[16];
#pragma unroll
  for (int f = 0; f < 16; ++f)
    qa[f] = *(const v2f*)(qp + (size_t)(q0 + l16) * 64 + f * 4 + hi * 2);

  const int nj = q0 / 16 + 1;  // key tiles needed (incl. diagonal tile)
  const float scale = 0.125f;  // 1/sqrt(64)
  const v8f zero = {0.f, 0.f, 0.f, 0.f, 0.f, 0.f, 0.f, 0.f};

  // ---- Pass 1: row max over causal keys ----
  float rmax[8];
#pragma unroll
  for (int r = 0; r < 8; ++r) rmax[r] = -3.0e38f;

  for (int j = 0; j < nj; ++j) {
    v8f s = zero;
#pragma unroll
    for (int f = 0; f < 16; ++f) {
      v2f bk = *(const v2f*)(kp + (size_t)(j * 16 + l16) * 64 + f * 4 + hi * 2);
      s = wmma_f32x4(qa[f], bk, s);
    }
    const int key = j * 16 + l16;
#pragma unroll
    for (int r = 0; r < 8; ++r) {
      const int qrow = q0 + r + hi * 8;
      const float sv = s[r] * scale;
      if (key <= qrow) rmax[r] = fmaxf(rmax[r], sv);
    }
  }
#pragma unroll
  for (int m = 1; m < 16; m <<= 1) {
#pragma unroll
    for (int r = 0; r < 8; ++r)
      rmax[r] = fmaxf(rmax[r], __shfl_xor(rmax[r], m, 32));
  }

  // ---- Pass 2: e = exp(s - max); stream to probs; rowsum; e @ V ----
  float rsum[8];
#pragma unroll
  for (int r = 0; r < 8; ++r) rsum[r] = 0.f;
  v8f oacc[4];
#pragma unroll
  for (int t = 0; t < 4; ++t) oacc[t] = zero;
  float* pw = pbuf[wave];

  for (int j = 0; j < nj; ++j) {
    v8f s = zero;
#pragma unroll
    for (int f = 0; f < 16; ++f) {
      v2f bk = *(const v2f*)(kp + (size_t)(j * 16 + l16) * 64 + f * 4 + hi * 2);
      s = wmma_f32x4(qa[f], bk, s);
    }
    const int key = j * 16 + l16;
#pragma unroll
    for (int r = 0; r < 8; ++r) {
      const int qrow = q0 + r + hi * 8;
      const float ev = (key <= qrow) ? __expf(s[r] * scale - rmax[r]) : 0.f;
      rsum[r] += ev;
      __builtin_nontemporal_store(ev, pp + (size_t)qrow * T + key);
      pw[(r + hi * 8) * 18 + l16] = ev;  // C-layout -> LDS
    }
    // LDS (A-layout) reads + V fragments + 16 WMMAs: oacc += P @ V
#pragma unroll
    for (int f = 0; f < 4; ++f) {        // f covers keys 4f..4f+3
      v2f pa = *(const v2f*)(pw + l16 * 18 + f * 4 + hi * 2);
      const int krow = j * 16 + f * 4 + hi * 2;
#pragma unroll
      for (int t = 0; t < 4; ++t) {      // t covers d = 16t..16t+15
        v2f bv;
        bv[0] = vp[(size_t)krow * 64 + t * 16 + l16];
        bv[1] = vp[(size_t)(krow + 1) * 64 + t * 16 + l16];
        oacc[t] = wmma_f32x4(pa, bv, oacc[t]);
      }
    }
  }

#pragma unroll
  for (int m = 1; m < 16; m <<= 1) {
#pragma unroll
    for (int r = 0; r < 8; ++r) rsum[r] += __shfl_xor(rsum[r], m, 32);
  }
  float inv[8];
#pragma unroll
  for (int r = 0; r < 8; ++r) inv[r] = 1.0f / rsum[r];

  // attn_out in (B, T, H*64) layout
#pragma unroll
  for (int t = 0; t < 4; ++t) {
#pragma unroll
    for (int r = 0; r < 8; ++r) {
      const int qrow = q0 + r + hi * 8;
      out[(size_t)(b * T + qrow) * (H * 64) + h * 64 + t * 16 + l16] =
          oacc[t][r] * inv[r];
    }
  }

  // ---- Pass 3: normalize the streamed probs in place (NT both ways) ----
  for (int j = 0; j < nj; ++j) {
    const int key = j * 16 + l16;
#pragma unroll
    for (int r = 0; r < 8; ++r) {
      const int qrow = q0 + r + hi * 8;
      float* addr = pp + (size_t)qrow * T + key;
      const float ev = __builtin_nontemporal_load(addr);
      __builtin_nontemporal_store(ev * inv[r], addr);  // masked are 0 -> 0
    }
  }
}

// ---------------------------------------------------------------------------
// Launch: x(2,2048,1024) f32, w_qkv(3072,1024) f32, w_out(1024,1024) f32.
// d_out = [output (4,194,304 f32) | attn_probs (134,217,728 f32)].
// ---------------------------------------------------------------------------
extern "C" void kernel_launch(void* const* d_in, const int* in_sizes, int n_in,
                              void* d_out, int out_size, void* d_ws,
                              size_t ws_size, hipStream_t stream) {
  const float* x     = (const float*)d_in[0];
  const float* w_qkv = (const float*)d_in[1];
  const float* w_out = (const float*)d_in[2];

  float* out   = (float*)d_out;
  float* probs = out + (size_t)4194304;  // 16MB offset -> 16B aligned

  float* ws   = (float*)d_ws;
  float* qkv  = ws;                       // 12,582,912 floats (48 MB)
  float* qw   = ws + (size_t)12582912;    //  4,194,304 floats (16 MB)
  float* kw   = qw + (size_t)4194304;
  float* vw   = kw + (size_t)4194304;
  float* attn = qkv;                      // alias: qkv dead after rope

  const int BT = 4096;   // B*T
  const int D  = 1024;
  const int T  = 2048;
  const int H  = 16;

  // 1) qkv = x @ w_qkv^T   (4096 x 3072 x 1024)
  gemm_nt_kernel<<<dim3(3072 / 64, BT / 64), 128, 0, stream>>>(
      x, w_qkv, qkv, BT, 3 * D, D);

  // 2) RoPE + split into (B,H,T,dk)
  rope_split_kernel<<<2097152 / 256, 256, 0, stream>>>(qkv, qw, kw, vw);

  // 3) zero attn_probs (upper triangle must be 0)
  zero_f4_kernel<<<(134217728 / 4) / 256, 256, 0, stream>>>((v4f*)probs);

  // 4) attention (writes probs + attn in (B,T,H*dk) layout)
  attn_kernel<<<dim3(T / 64, 2 * H), 128, 0, stream>>>(
      qw, kw, vw, probs, attn, T, H);

  // 5) output = attn @ w_out^T   (4096 x 1024 x 1024)
  gemm_nt_kernel<<<dim3(D / 64, BT / 64), 128, 0, stream>>>(
      attn, w_out, out, BT, D, D);
}